// S4Recurrent_50096498540855
// MI455X (gfx1250) — compile-verified
//
#include <hip/hip_runtime.h>
#include <hip/hip_bf16.h>

typedef __attribute__((ext_vector_type(2))) float v2f;
typedef __attribute__((ext_vector_type(8))) float v8f;

#define D_MODEL 1024
#define D_STATE 64
#define SEQ     4096
#define BATCH   8

// ---------------------------------------------------------------------------
// Kernel 1: setup. One block, 1024 threads.
// Computes K[0..4095] = C * A_bar^j * B_bar  (+D folded into K[0]).
// ---------------------------------------------------------------------------
__device__ __forceinline__ void mm64(const float* Am, const float* Bm, float* Cm, int tid) {
    // C = A @ B, 64x64. thread: row r = tid>>4, cols j0..j0+3
    int r = tid >> 4;
    int j0 = (tid & 15) * 4;
    float c0 = 0.f, c1 = 0.f, c2 = 0.f, c3 = 0.f;
    for (int k = 0; k < 64; ++k) {
        float a = Am[r * 64 + k];
        const float* br = &Bm[k * 64 + j0];
        c0 = fmaf(a, br[0], c0);
        c1 = fmaf(a, br[1], c1);
        c2 = fmaf(a, br[2], c2);
        c3 = fmaf(a, br[3], c3);
    }
    float* cr = &Cm[r * 64 + j0];
    cr[0] = c0; cr[1] = c1; cr[2] = c2; cr[3] = c3;
}

__device__ __forceinline__ float ident_at(int i) {
    return ((i >> 6) == (i & 63)) ? 1.f : 0.f;
}

__global__ __launch_bounds__(1024) void s4_setup_kernel(
    const float* __restrict__ A, const float* __restrict__ Bm,
    const float* __restrict__ Cv, const float* __restrict__ Dv,
    const float* __restrict__ log_dt, float* __restrict__ Kout) {
    __shared__ float Z[64 * 64];     // dt*A, later A_bar^64
    __shared__ float ABAR[64 * 64];  // expm(dt*A)
    __shared__ float PHI[64 * 64];   // phi1, later squaring scratch, later S[c][r]
    __shared__ float TMP[64 * 64];   // scratch, later P[r][j]
    __shared__ float Bbar[64];

    const int tid = threadIdx.x;
    float dt = expf(log_dt[0]);
    dt = fminf(fmaxf(dt, 0.001f), 0.1f);

    // Z = dt * A
    for (int i = tid; i < 4096; i += 1024) Z[i] = dt * A[i];
    __syncthreads();

    // ABAR = expm(Z), Taylor-Horner, 12 terms: R = I + Z/12; R = I + (Z@R)/t
    for (int i = tid; i < 4096; i += 1024) ABAR[i] = ident_at(i) + Z[i] * (1.f / 12.f);
    __syncthreads();
    for (int t = 11; t >= 1; --t) {
        mm64(Z, ABAR, TMP, tid);
        __syncthreads();
        float inv = 1.f / (float)t;
        for (int i = tid; i < 4096; i += 1024) ABAR[i] = ident_at(i) + TMP[i] * inv;
        __syncthreads();
    }

    // PHI = phi1(Z) = sum Z^k/(k+1)!  (inv(Z)*(expm(Z)-I), no matrix inverse needed)
    for (int i = tid; i < 4096; i += 1024) PHI[i] = ident_at(i) + Z[i] * (1.f / 11.f);
    __syncthreads();
    for (int t = 10; t >= 2; --t) {
        mm64(Z, PHI, TMP, tid);
        __syncthreads();
        float inv = 1.f / (float)t;
        for (int i = tid; i < 4096; i += 1024) PHI[i] = ident_at(i) + TMP[i] * inv;
        __syncthreads();
    }

    // B_bar = PHI @ (dt * Bm)
    if (tid < 64) {
        float s = 0.f;
        for (int k = 0; k < 64; ++k) s = fmaf(PHI[tid * 64 + k], Bm[k], s);
        Bbar[tid] = dt * s;
    }
    __syncthreads();

    // M64 = ABAR^64 via 6 squarings (final result lands in Z; Z no longer needed)
    mm64(ABAR, ABAR, TMP, tid); __syncthreads(); // A^2
    mm64(TMP, TMP, PHI, tid);  __syncthreads();  // A^4
    mm64(PHI, PHI, TMP, tid);  __syncthreads();  // A^8
    mm64(TMP, TMP, PHI, tid);  __syncthreads();  // A^16
    mm64(PHI, PHI, TMP, tid);  __syncthreads();  // A^32
    mm64(TMP, TMP, Z, tid);    __syncthreads();  // A^64 -> Z

    // S[c][:] = M64^c @ Bbar  (chunk-start state vectors), stored in PHI
    if (tid < 64) PHI[tid] = Bbar[tid];
    __syncthreads();
    for (int c = 1; c < 64; ++c) {
        if (tid < 64) {
            float v = 0.f;
            for (int k = 0; k < 64; ++k) v = fmaf(Z[tid * 64 + k], PHI[(c - 1) * 64 + k], v);
            PHI[c * 64 + tid] = v;
        }
        __syncthreads();
    }

    // P[r][:] = C @ ABAR^r  (row vectors), stored in TMP
    if (tid < 64) TMP[tid] = Cv[tid];
    __syncthreads();
    for (int r = 1; r < 64; ++r) {
        if (tid < 64) {
            float v = 0.f;
            for (int i = 0; i < 64; ++i) v = fmaf(TMP[(r - 1) * 64 + i], ABAR[i * 64 + tid], v);
            TMP[r * 64 + tid] = v;
        }
        __syncthreads();
    }

    // K[64c + r] = dot(P[r], S[c]);  K[0] += D
    float Dval = Dv[0];
    for (int q = 0; q < 4; ++q) {
        int j = tid + q * 1024;
        int c = j >> 6, r = j & 63;
        float s = 0.f;
        for (int k = 0; k < 64; ++k) s = fmaf(TMP[r * 64 + k], PHI[c * 64 + k], s);
        Kout[j] = s + ((j == 0) ? Dval : 0.f);
    }
}

// ---------------------------------------------------------------------------
// Kernel 2: input projection. One wave per (b,t) row; streams 128 MB of u.
// Up[t*8 + b] = dot(u[b,t,:], W_in)
// ---------------------------------------------------------------------------
__global__ __launch_bounds__(256) void s4_proj_kernel(
    const float* __restrict__ u, const float* __restrict__ Win,
    float* __restrict__ Up) {
    int wave = threadIdx.x >> 5;
    int lane = threadIdx.x & 31;
    int row = blockIdx.x * 8 + wave;  // row = b*SEQ + t, 0..32767
    const float4* u4 = (const float4*)(u + (size_t)row * D_MODEL);
    const float4* w4 = (const float4*)Win;
    float s = 0.f;
#pragma unroll
    for (int i = 0; i < 8; ++i) {
        float4 a = u4[lane + i * 32];
        float4 w = w4[lane + i * 32];
        s = fmaf(a.x, w.x, s);
        s = fmaf(a.y, w.y, s);
        s = fmaf(a.z, w.z, s);
        s = fmaf(a.w, w.w, s);
    }
#pragma unroll
    for (int off = 16; off > 0; off >>= 1) s += __shfl_xor(s, off, 32);
    if (lane == 0) {
        int b = row >> 12;       // row / SEQ
        int t = row & (SEQ - 1); // row % SEQ
        Up[t * BATCH + b] = s;
    }
}

// ---------------------------------------------------------------------------
// Kernel 3: causal Toeplitz matmul Y = T @ Up via V_WMMA_F32_16X16X4_F32.
// 256 tiles (16 t-values x 16 cols: 8 batches + zero pad), 32 blocks x 8 waves,
// tile = wave*32 + block for load balance. All LDS loads are UNCONDITIONAL:
//  - K is front-padded with 16 zeros so acausal taps read zero
//  - lanes n>=8 aim their B pointer at a 128-float zero region with stride 0
// 4 independent accumulators / 4 WMMAs per iteration break the RAW chain.
// ---------------------------------------------------------------------------
#define WMMA_F32(a, b, c) \
    __builtin_amdgcn_wmma_f32_16x16x4_f32(false, (a), false, (b), (short)0, (c), false, false)

__global__ __launch_bounds__(256) void s4_conv_kernel(
    const float* __restrict__ Kker, const float* __restrict__ Up,
    float* __restrict__ Yt) {
    extern __shared__ float smem[];
    float* Kpad = smem;                    // [16 + 4096] front zero pad + taps
    float* Upl  = smem + 4112;             // [4096 * 8]
    float* Zreg = smem + 4112 + SEQ * BATCH; // [128] zeros for n>=8 lanes

    // cooperative fill
    if (threadIdx.x < 16) Kpad[threadIdx.x] = 0.f;
    if (threadIdx.x >= 128 && threadIdx.x < 256) Zreg[threadIdx.x - 128] = 0.f;
    for (int i = threadIdx.x; i < SEQ; i += 256) Kpad[16 + i] = Kker[i];
    {
        float4* dst = (float4*)Upl;
        const float4* src = (const float4*)Up;
        for (int i = threadIdx.x; i < (SEQ * BATCH) / 4; i += 256) dst[i] = src[i];
    }
    __syncthreads();

    int wave = threadIdx.x >> 5;
    int lane = threadIdx.x & 31;
    int tile = wave * 32 + blockIdx.x;     // 0..255, balanced across blocks
    int t0 = tile * 16;
    int m = lane & 15;                     // A row M / B,D col N
    int half = lane >> 4;
    int k0 = half * 2;                     // K-slot pair owned by this half-wave
    bool valid = (m < BATCH);

    // A base: lowest logical tap index touched in an iteration is (t0+m-k0-13);
    // +16 for the zero pad => all DS offsets are immediates in {0,1,4,5,8,9,12,13}.
    const float* aB = Kpad + (t0 + m - k0 + 3);
    const float* bB = valid ? (Upl + (k0 * BATCH + m)) : Zreg;
    int bStep = valid ? (16 * BATCH) : 0;  // 16 s-values per iteration

    v8f acc0 = {0.f,0.f,0.f,0.f,0.f,0.f,0.f,0.f};
    v8f acc1 = acc0, acc2 = acc0, acc3 = acc0;

    int iters = tile + 1;                  // (t0/4 + 4) / 4, exact
    for (int it = 0; it < iters; ++it) {
        v2f a0, a1, a2, a3, b0, b1, b2, b3;
        a0.x = aB[13]; a0.y = aB[12];      // s4 = 4*it + 0
        a1.x = aB[9];  a1.y = aB[8];       // s4 = 4*it + 1
        a2.x = aB[5];  a2.y = aB[4];       // s4 = 4*it + 2
        a3.x = aB[1];  a3.y = aB[0];       // s4 = 4*it + 3
        b0.x = bB[0];   b0.y = bB[8];
        b1.x = bB[32];  b1.y = bB[40];
        b2.x = bB[64];  b2.y = bB[72];
        b3.x = bB[96];  b3.y = bB[104];
        acc0 = WMMA_F32(a0, b0, acc0);
        acc1 = WMMA_F32(a1, b1, acc1);
        acc2 = WMMA_F32(a2, b2, acc2);
        acc3 = WMMA_F32(a3, b3, acc3);
        aB -= 16;
        bB += bStep;
    }

    if (valid) {
#pragma unroll
        for (int i = 0; i < 8; ++i) {
            int t = t0 + half * 8 + i;     // D layout: VGPR i holds rows i / i+8
            Yt[t * BATCH + m] = (acc0[i] + acc1[i]) + (acc2[i] + acc3[i]);
        }
    }
}

// ---------------------------------------------------------------------------
// Kernel 4: output expansion. out[b,t,:] = Yt[t,b] * W_out. Streams 128 MB.
// ---------------------------------------------------------------------------
__global__ __launch_bounds__(256) void s4_expand_kernel(
    const float* __restrict__ Yt, const float* __restrict__ Wout,
    float* __restrict__ out) {
    int row = blockIdx.x;            // b*SEQ + t
    int b = row >> 12;
    int t = row & (SEQ - 1);
    float y = Yt[t * BATCH + b];
    float4 w = ((const float4*)Wout)[threadIdx.x];
    float4 o;
    o.x = y * w.x; o.y = y * w.y; o.z = y * w.z; o.w = y * w.w;
    ((float4*)out)[(size_t)row * (D_MODEL / 4) + threadIdx.x] = o;
}

// ---------------------------------------------------------------------------
extern "C" void kernel_launch(void* const* d_in, const int* in_sizes, int n_in,
                              void* d_out, int out_size, void* d_ws, size_t ws_size,
                              hipStream_t stream) {
    (void)in_sizes; (void)n_in; (void)out_size; (void)ws_size;
    const float* u      = (const float*)d_in[0];  // [8,4096,1024]
    const float* A      = (const float*)d_in[1];  // [64,64]
    const float* Bm     = (const float*)d_in[2];  // [64,1]
    const float* Cv     = (const float*)d_in[3];  // [1,64]
    const float* Dv     = (const float*)d_in[4];  // [1]
    const float* log_dt = (const float*)d_in[5];  // [1]
    const float* Win    = (const float*)d_in[6];  // [1,1024]
    const float* Wout   = (const float*)d_in[7];  // [1024,1]
    float* out = (float*)d_out;

    float* ws = (float*)d_ws;
    float* Kker = ws;                 // 4096
    float* Up   = ws + 4096;          // 4096*8
    float* Yt   = Up + SEQ * BATCH;   // 4096*8

    s4_setup_kernel<<<1, 1024, 0, stream>>>(A, Bm, Cv, Dv, log_dt, Kker);
    s4_proj_kernel<<<(BATCH * SEQ) / 8, 256, 0, stream>>>(u, Win, Up);
    size_t conv_lds = (4112 + SEQ * BATCH + 128) * sizeof(float);  // ~144.7 KB
    s4_conv_kernel<<<32, 256, conv_lds, stream>>>(Kker, Up, Yt);
    s4_expand_kernel<<<BATCH * SEQ, 256, 0, stream>>>(Yt, Wout, out);
}